// PointNetEncoder_27590869909776
// MI455X (gfx1250) — compile-verified
//
#include <hip/hip_runtime.h>
#include <cstddef>
#include <cstdint>

// ---------------------------------------------------------------------------
// PointNet++ (MSG) encoder for gfx1250. All GEMM stages use
// V_WMMA_F32_16X16X4_F32 (fp32 WMMA) -- the network is bandwidth-bound
// (~30 GFLOP vs ~1-2 GB traffic @ 23.3 TB/s), so fp32 fidelity is free.
// Weight tiles are staged global->LDS with GLOBAL_LOAD_ASYNC_TO_LDS_B32
// (ASYNCcnt) and each wave computes NT adjacent 16x16 output tiles so one
// A-fragment load feeds NT WMMAs (4x less X traffic).
// ---------------------------------------------------------------------------

typedef float v2f_t __attribute__((ext_vector_type(2)));
typedef float v8f_t __attribute__((ext_vector_type(8)));

static __device__ __forceinline__ unsigned wave_ballot32(bool p) {
#if __has_builtin(__builtin_amdgcn_ballot_w32)
  return __builtin_amdgcn_ballot_w32(p);
#else
  return (unsigned)__ballot(p);
#endif
}

// ------------------------- (B,3,N) -> (B,N,3) ------------------------------
__global__ void k_transpose(const float* __restrict__ in, float* __restrict__ out,
                            int B, int N) {
  size_t tot = (size_t)B * N * 3;
  for (size_t i = (size_t)blockIdx.x * blockDim.x + threadIdx.x; i < tot;
       i += (size_t)gridDim.x * blockDim.x) {
    size_t b = i / ((size_t)N * 3);
    size_t rem = i % ((size_t)N * 3);
    int n = (int)(rem / 3), d = (int)(rem % 3);
    out[i] = in[(b * 3 + d) * (size_t)N + n];
  }
}

// ------------------------- farthest point sampling -------------------------
// One block per batch. xyz (SoA) + running min-dist live in LDS.
__global__ void k_fps(const float* __restrict__ pts, int* __restrict__ outIdx,
                      int N, int npoint) {
  extern __shared__ float sm[];
  float* xs = sm;
  float* ys = xs + N;
  float* zs = ys + N;
  float* dist = zs + N;
  float* redv = dist + N;           // 256 floats
  int*   redi = (int*)(redv + 256); // 256 ints
  int b = blockIdx.x, tid = threadIdx.x, bs = blockDim.x;
  const float* base = pts + (size_t)b * N * 3;
  for (int j = tid; j < N; j += bs) {
    xs[j] = base[j * 3 + 0];
    ys[j] = base[j * 3 + 1];
    zs[j] = base[j * 3 + 2];
    dist[j] = 1e10f;
  }
  __syncthreads();
  int far = 0;
  for (int it = 0; it < npoint; ++it) {
    if (tid == 0) outIdx[(size_t)b * npoint + it] = far;  // emit incoming carry
    float cx = xs[far], cy = ys[far], cz = zs[far];
    float bv = -1.f; int bi = 0;
    for (int j = tid; j < N; j += bs) {
      float dx = xs[j] - cx, dy = ys[j] - cy, dz = zs[j] - cz;
      float d = dx * dx + dy * dy + dz * dz;
      float dm = fminf(dist[j], d);
      dist[j] = dm;
      if (dm > bv) { bv = dm; bi = j; }
    }
    redv[tid] = bv; redi[tid] = bi;
    __syncthreads();
    for (int w = bs >> 1; w > 0; w >>= 1) {
      if (tid < w) {
        float ov = redv[tid + w]; int oi = redi[tid + w];
        if (ov > redv[tid] || (ov == redv[tid] && oi < redi[tid])) {
          redv[tid] = ov; redi[tid] = oi;
        }
      }
      __syncthreads();
    }
    far = redi[0];
    __syncthreads();
  }
}

// ------------------------- gather sampled coordinates ----------------------
__global__ void k_gather_xyz(const float* __restrict__ pts, const int* __restrict__ idx,
                             float* __restrict__ out, int B, int N, int S) {
  size_t tot = (size_t)B * S * 3;
  for (size_t i = (size_t)blockIdx.x * blockDim.x + threadIdx.x; i < tot;
       i += (size_t)gridDim.x * blockDim.x) {
    size_t b = i / ((size_t)S * 3);
    size_t rem = i % ((size_t)S * 3);
    int s = (int)(rem / 3), d = (int)(rem % 3);
    out[i] = pts[((size_t)b * N + idx[b * S + s]) * 3 + d];
  }
}

// ------------------------- ball query + grouping ---------------------------
// One block per (b,s) center. Wave 0 compacts the first K in-radius indices
// (ascending order == jnp.sort semantics) via wave32 ballot, then the whole
// block gathers the grouped features [feats | rel_xyz | 0-pad] into X.
__global__ void k_ball_group(const float* __restrict__ xyz,
                             const float* __restrict__ feats, int Cf,
                             const float* __restrict__ newxyz,
                             float* __restrict__ X,
                             int N, int S, int K, int cinPad, float r2) {
  __shared__ int sidx[128];
  __shared__ int scount;
  int blk = blockIdx.x;
  int b = blk / S, s = blk % S;
  const float* c = newxyz + (size_t)(b * S + s) * 3;
  float cx = c[0], cy = c[1], cz = c[2];
  int tid = threadIdx.x;
  if (tid < 32) {
    int count = 0;
    for (int base = 0; base < N && count < K; base += 32) {
      int j = base + tid;
      bool inb = false;
      if (j < N) {
        const float* p = xyz + ((size_t)b * N + j) * 3;
        float dx = p[0] - cx, dy = p[1] - cy, dz = p[2] - cz;
        inb = (dx * dx + dy * dy + dz * dz) <= r2;
      }
      unsigned m = wave_ballot32(inb);
      while (m && count < K) {
        int bit = __builtin_ctz(m);
        if (tid == 0) sidx[count] = base + bit;
        count++;
        m &= m - 1;
      }
    }
    if (tid == 0) scount = count;
  }
  __syncthreads();
  int cnt = scount;
  for (int k = tid; k < K; k += blockDim.x)
    if (k >= cnt) sidx[k] = (cnt > 0) ? sidx[0] : 0;  // pad with first index
  __syncthreads();
  size_t rowBase = (size_t)(b * S + s) * K;
  int tot = K * cinPad;
  for (int t = tid; t < tot; t += blockDim.x) {
    int k = t / cinPad, cc = t % cinPad;
    int j = sidx[k];
    float v = 0.f;
    if (cc < Cf) {
      v = feats[((size_t)b * N + j) * Cf + cc];
    } else if (cc < Cf + 3) {
      const float* p = xyz + ((size_t)b * N + j) * 3;
      int d = cc - Cf;
      v = p[d] - ((d == 0) ? cx : (d == 1) ? cy : cz);
    }
    X[(rowBase + k) * (size_t)cinPad + cc] = v;
  }
}

// ------------------------- WMMA fp32 GEMM: Y = X * W^T + b -----------------
// X: R x cinPad (row-major, zero-padded K), W: Cout x Cin (dense, from d_in),
// Y: R x ystride (ystride = 16-aligned; pad cols written as 0).
// Each wave computes NT adjacent 16x16 output tiles: one A fragment load
// feeds NT WMMAs. Weight tile async-copied to LDS (ASYNCcnt path).
template <int NT>
__global__ void k_gemm_bias(const float* __restrict__ X, const float* __restrict__ W,
                            const float* __restrict__ bias, float* __restrict__ Y,
                            int R, int Cin, int cinPad, int Cout, int ystride) {
  extern __shared__ float wl[];
  const int lstr = cinPad + 2;   // LDS row stride (bank-conflict pad, 8B aligned)
  const int COLS = 16 * NT;
  int o0 = blockIdx.y * COLS;

  // 1) zero the padded region (disjoint from the async-loaded region below,
  //    so DScnt stores and ASYNCcnt loads cannot race on an address).
  for (int t = threadIdx.x; t < COLS * cinPad; t += blockDim.x) {
    int oo = t / cinPad, kk = t % cinPad;
    if (!((o0 + oo < Cout) && (kk < Cin))) wl[oo * lstr + kk] = 0.f;
  }
  // 2) async-copy the valid weight tile global->LDS (GLOBAL_LOAD_ASYNC_TO_LDS).
  for (int t = threadIdx.x; t < COLS * Cin; t += blockDim.x) {
    int oo = t / Cin, kk = t % Cin;
    if (o0 + oo < Cout) {
      unsigned ldsa = (unsigned)(uintptr_t)&wl[oo * lstr + kk];
      unsigned long long ga =
          (unsigned long long)(uintptr_t)&W[(size_t)(o0 + oo) * Cin + kk];
      asm volatile("global_load_async_to_lds_b32 %0, %1, off"
                   :: "v"(ldsa), "v"(ga) : "memory");
    }
  }
  asm volatile("s_wait_asynccnt 0x0" ::: "memory");
  __syncthreads();

  int wave = threadIdx.x >> 5, lane = threadIdx.x & 31;
  int half = lane >> 4, l16 = lane & 15;
  int rowTiles = (R + 15) >> 4;
  int rtStride = gridDim.x * 8;
  const v8f_t vzero = {0.f, 0.f, 0.f, 0.f, 0.f, 0.f, 0.f, 0.f};
  for (int rt = blockIdx.x * 8 + wave; rt < rowTiles; rt += rtStride) {
    int rowA = rt * 16 + l16;
    int rowS = rowA < R ? rowA : (R - 1);   // clamp; tail rows never stored
    const float* xr = X + (size_t)rowS * cinPad;
    // prefetch this lane's row in the wave's next row tile
    int nr = rowA + rtStride * 16;
    if (nr < R) __builtin_prefetch(X + (size_t)nr * cinPad, 0, 1);
    v8f_t acc[NT];
#pragma unroll
    for (int t = 0; t < NT; ++t) acc[t] = vzero;
    for (int k = 0; k < cinPad; k += 4) {
      int ka = k + half * 2;
      // A 16x4 f32: lanes 0-15 -> K=k..k+1, lanes 16-31 -> K=k+2..k+3 (M=l16)
      v2f_t a; a.x = xr[ka]; a.y = xr[ka + 1];
#pragma unroll
      for (int t = 0; t < NT; ++t) {
        // B 4x16 f32: same K striping, N = l16 (col o0 + t*16 + l16)
        const float* wr = wl + (size_t)(t * 16 + l16) * lstr;
        v2f_t bb; bb.x = wr[ka]; bb.y = wr[ka + 1];
        acc[t] = __builtin_amdgcn_wmma_f32_16x16x4_f32(false, a, false, bb,
                                                       (short)0, acc[t],
                                                       false, false);
      }
    }
#pragma unroll
    for (int t = 0; t < NT; ++t) {
      int o = o0 + t * 16 + l16;
      if (o < ystride) {
        float bval = (o < Cout) ? bias[o] : 0.f;
        for (int v = 0; v < 8; ++v) {
          int r = rt * 16 + v + half * 8;  // C/D: VGPR v -> M=v (+8 hi half)
          if (r < R) {
            float val = (o < Cout) ? (acc[t][v] + bval) : 0.f;  // pad cols = 0
            Y[(size_t)r * ystride + o] = val;
          }
        }
      }
    }
  }
}

// ------------------------- per-channel sum / sumsq -------------------------
__global__ void k_colstats(const float* __restrict__ Y, int R, int C, int stride,
                           float* __restrict__ stats) {
  __shared__ float sh[256], sh2[256];
  int c = blockIdx.y;
  float s = 0.f, sq = 0.f;
  for (int r = blockIdx.x * blockDim.x + threadIdx.x; r < R;
       r += gridDim.x * blockDim.x) {
    float v = Y[(size_t)r * stride + c];
    s += v; sq += v * v;
  }
  sh[threadIdx.x] = s; sh2[threadIdx.x] = sq;
  __syncthreads();
  for (int w = 128; w > 0; w >>= 1) {
    if ((int)threadIdx.x < w) {
      sh[threadIdx.x] += sh[threadIdx.x + w];
      sh2[threadIdx.x] += sh2[threadIdx.x + w];
    }
    __syncthreads();
  }
  if (threadIdx.x == 0) {
    atomicAdd(&stats[c], sh[0]);
    atomicAdd(&stats[C + c], sh2[0]);
  }
}

// ------------------------- batchnorm (global stats) + ReLU -----------------
__global__ void k_bn_relu(float* __restrict__ Y, int R, int C, int stride,
                          const float* __restrict__ stats,
                          const float* __restrict__ g,
                          const float* __restrict__ be) {
  size_t tot = (size_t)R * C;
  float invR = 1.0f / (float)R;
  for (size_t i = (size_t)blockIdx.x * blockDim.x + threadIdx.x; i < tot;
       i += (size_t)gridDim.x * blockDim.x) {
    size_t r = i / C; int c = (int)(i % C);
    float m = stats[c] * invR;
    float v = stats[C + c] * invR - m * m;
    float x = Y[r * stride + c];
    x = g[c] * (x - m) * rsqrtf(v + 1e-5f) + be[c];
    Y[r * stride + c] = x > 0.f ? x : 0.f;
  }
}

// ------------------------- max over K samples per group --------------------
__global__ void k_maxpool(const float* __restrict__ in, float* __restrict__ out,
                          int G, int K, int C, int inStride, int outStride,
                          int outOffset) {
  size_t tot = (size_t)G * C;
  for (size_t i = (size_t)blockIdx.x * blockDim.x + threadIdx.x; i < tot;
       i += (size_t)gridDim.x * blockDim.x) {
    size_t gidx = i / C; int c = (int)(i % C);
    const float* p = in + (gidx * K) * (size_t)inStride + c;
    float m = -3.4e38f;
    for (int k = 0; k < K; ++k) m = fmaxf(m, p[(size_t)k * inStride]);
    out[gidx * (size_t)outStride + outOffset + c] = m;
  }
}

// ------------------------- group-all staging: [xyz | points] ---------------
__global__ void k_groupall(const float* __restrict__ xyz, const float* __restrict__ pts,
                           float* __restrict__ X, int G, int Cf, int cinPad) {
  size_t tot = (size_t)G * cinPad;
  for (size_t i = (size_t)blockIdx.x * blockDim.x + threadIdx.x; i < tot;
       i += (size_t)gridDim.x * blockDim.x) {
    size_t row = i / cinPad; int c = (int)(i % cinPad);
    float v = 0.f;
    if (c < 3) v = xyz[row * 3 + c];
    else if (c < 3 + Cf) v = pts[row * (size_t)Cf + (c - 3)];
    X[i] = v;
  }
}

// ------------------------- fp3 staging: [l2_points | broadcast(l3)] --------
__global__ void k_fp3_stage(const float* __restrict__ l2p, const float* __restrict__ l3p,
                            float* __restrict__ X, int G, int Srows) {
  size_t tot = (size_t)G * 1536;
  for (size_t i = (size_t)blockIdx.x * blockDim.x + threadIdx.x; i < tot;
       i += (size_t)gridDim.x * blockDim.x) {
    size_t row = i / 1536; int c = (int)(i % 1536);
    size_t b = row / Srows;
    X[i] = (c < 512) ? l2p[row * 512 + c] : l3p[b * 1024 + (c - 512)];
  }
}

// ------------------------- 3-NN with inverse-distance weights --------------
__global__ void k_knn3(const float* __restrict__ xyz1, const float* __restrict__ xyz2,
                       int* __restrict__ idx3, float* __restrict__ w3,
                       int Btot, int N, int S) {
  int gid = blockIdx.x * blockDim.x + threadIdx.x;
  if (gid >= Btot) return;
  int b = gid / N;
  const float* p = xyz1 + (size_t)gid * 3;
  float px = p[0], py = p[1], pz = p[2];
  float d0 = 3.4e38f, d1 = 3.4e38f, d2 = 3.4e38f;
  int i0 = 0, i1 = 0, i2 = 0;
  const float* q = xyz2 + (size_t)b * S * 3;
  for (int s = 0; s < S; ++s) {
    float dx = q[s * 3] - px, dy = q[s * 3 + 1] - py, dz = q[s * 3 + 2] - pz;
    float d = dx * dx + dy * dy + dz * dz;
    if (d < d0)      { d2 = d1; i2 = i1; d1 = d0; i1 = i0; d0 = d; i0 = s; }
    else if (d < d1) { d2 = d1; i2 = i1; d1 = d;  i1 = s; }
    else if (d < d2) { d2 = d;  i2 = s; }
  }
  float r0 = 1.f / (d0 + 1e-8f), r1 = 1.f / (d1 + 1e-8f), r2 = 1.f / (d2 + 1e-8f);
  float t = r0 + r1 + r2;
  w3[(size_t)gid * 3 + 0] = r0 / t;
  w3[(size_t)gid * 3 + 1] = r1 / t;
  w3[(size_t)gid * 3 + 2] = r2 / t;
  idx3[(size_t)gid * 3 + 0] = i0;
  idx3[(size_t)gid * 3 + 1] = i1;
  idx3[(size_t)gid * 3 + 2] = i2;
}

// ------------------------- fp staging: [points1 | interp] ------------------
__global__ void k_fp_stage(const float* __restrict__ p1, int c1, int s1,
                           const float* __restrict__ p2, int c2, int s2,
                           const int* __restrict__ idx3, const float* __restrict__ w3,
                           float* __restrict__ X, int G, int N, int S, int cinPad) {
  size_t tot = (size_t)G * cinPad;
  for (size_t i = (size_t)blockIdx.x * blockDim.x + threadIdx.x; i < tot;
       i += (size_t)gridDim.x * blockDim.x) {
    size_t row = i / cinPad; int c = (int)(i % cinPad);
    size_t b = row / N;
    float v = 0.f;
    if (c < c1) {
      v = p1[row * (size_t)s1 + c];
    } else if (c < c1 + c2) {
      int cc = c - c1;
      const int* id = idx3 + row * 3;
      const float* w = w3 + row * 3;
      v = w[0] * p2[(b * S + id[0]) * (size_t)s2 + cc]
        + w[1] * p2[(b * S + id[1]) * (size_t)s2 + cc]
        + w[2] * p2[(b * S + id[2]) * (size_t)s2 + cc];
    }
    X[i] = v;
  }
}

// ===========================================================================
// Host orchestration
// ===========================================================================

struct LayerP { const float* W; const float* b; const float* g; const float* be; int cout; };

static inline int ru16(int x) { return (x + 15) & ~15; }
static inline int gblocks(size_t tot, int cap = 32768) {
  long long b = (long long)((tot + 255) / 256);
  if (b > cap) b = cap;
  if (b < 1) b = 1;
  return (int)b;
}

static void launch_gemm(const float* X, const LayerP& L, float* Y, int R,
                        int cin, int cinPad, int ystride, hipStream_t st) {
  int rowTiles = (R + 15) / 16;
  int colTiles = ru16(L.cout) / 16;
  int gx = (rowTiles + 7) / 8;
  if (cinPad <= 1024) {
    int gy = (colTiles + 3) / 4;
    size_t lds = (size_t)64 * (cinPad + 2) * sizeof(float);
    k_gemm_bias<4><<<dim3(gx, gy), 256, lds, st>>>(X, L.W, L.b, Y, R, cin,
                                                   cinPad, L.cout, ystride);
  } else {
    int gy = (colTiles + 1) / 2;
    size_t lds = (size_t)32 * (cinPad + 2) * sizeof(float);
    k_gemm_bias<2><<<dim3(gx, gy), 256, lds, st>>>(X, L.W, L.b, Y, R, cin,
                                                   cinPad, L.cout, ystride);
  }
}

static void launch_bn(float* Y, int R, int C, int stride, const float* g,
                      const float* be, float* stats, hipStream_t st) {
  hipMemsetAsync(stats, 0, (size_t)2 * C * sizeof(float), st);
  int gx = (R + 255) / 256; if (gx > 256) gx = 256;
  k_colstats<<<dim3(gx, C), 256, 0, st>>>(Y, R, C, stride, stats);
  k_bn_relu<<<gblocks((size_t)R * C), 256, 0, st>>>(Y, R, C, stride, stats, g, be);
}

// Chain of conv+BN+ReLU layers; input X lives in `pong` region; intermediates
// ping-pong (layer0 -> ping, layer1 -> pong, ...), last layer -> finalDst.
static void run_chain(const float* X, int R, int cin, int cinPad,
                      const LayerP* Ls, int nL, float* ping, float* pong,
                      float* finalDst, int finalStride, float* stats,
                      hipStream_t st) {
  const float* cur = X; int ccin = cin, cpad = cinPad;
  for (int l = 0; l < nL; ++l) {
    bool last = (l == nL - 1);
    float* dst = last ? finalDst : ((l & 1) ? pong : ping);
    int ys = last ? finalStride : ru16(Ls[l].cout);
    launch_gemm(cur, Ls[l], dst, R, ccin, cpad, ys, st);
    launch_bn(dst, R, Ls[l].cout, ys, Ls[l].g, Ls[l].be, stats, st);
    cur = dst; ccin = Ls[l].cout; cpad = ys;
  }
}

extern "C" void kernel_launch(void* const* d_in, const int* in_sizes, int n_in,
                              void* d_out, int out_size, void* d_ws, size_t ws_size,
                              hipStream_t stream) {
  (void)in_sizes; (void)n_in; (void)out_size; (void)ws_size;
  const int B = 8, N0 = 4096;
  const float* xyz_in = (const float*)d_in[0];

  // ---- parameter extraction (depth-first insertion order of setup_inputs) --
  int pi = 1;
  auto nx = [&]() -> const float* { return (const float*)d_in[pi++]; };
  auto takeL = [&](int cout) -> LayerP {
    LayerP L; L.W = nx(); L.b = nx(); L.g = nx(); L.be = nx(); L.cout = cout; return L;
  };
  LayerP sa1b[3][3];
  const int sa1c[3][3] = {{32, 32, 64}, {64, 64, 128}, {64, 96, 128}};
  for (int i = 0; i < 3; ++i) for (int j = 0; j < 3; ++j) sa1b[i][j] = takeL(sa1c[i][j]);
  LayerP sa2b[2][3];
  const int sa2c[2][3] = {{128, 128, 256}, {128, 196, 256}};
  for (int i = 0; i < 2; ++i) for (int j = 0; j < 3; ++j) sa2b[i][j] = takeL(sa2c[i][j]);
  LayerP sa3L[3]; const int sa3c[3] = {256, 512, 1024};
  for (int j = 0; j < 3; ++j) sa3L[j] = takeL(sa3c[j]);
  LayerP fp3L[2]; fp3L[0] = takeL(256); fp3L[1] = takeL(256);
  LayerP fp2L[2]; fp2L[0] = takeL(256); fp2L[1] = takeL(128);
  LayerP fp1L[2]; fp1L[0] = takeL(128); fp1L[1] = takeL(128);
  const float* W1 = nx(); const float* b1 = nx();
  const float* g1 = nx(); const float* be1 = nx();
  const float* W2 = nx(); const float* b2 = nx();

  // ---- workspace layout ----------------------------------------------------
  float* ws = (float*)d_ws;
  size_t off = 0;
  auto ALLOC = [&](size_t n) -> float* {
    float* p = ws + off; off += (n + 15) & ~(size_t)15; return p;
  };
  float* p0     = ALLOC((size_t)B * N0 * 3);
  int*   fps1   = (int*)ALLOC((size_t)B * 512);
  float* l1xyz  = ALLOC((size_t)B * 512 * 3);
  int*   fps2   = (int*)ALLOC((size_t)B * 128);
  float* l2xyz  = ALLOC((size_t)B * 128 * 3);
  float* l1pts  = ALLOC((size_t)B * 512 * 320);
  float* l2pts  = ALLOC((size_t)B * 128 * 512);
  float* l3pts  = ALLOC((size_t)B * 1024);
  float* l2new  = ALLOC((size_t)B * 128 * 256);
  float* l1new  = ALLOC((size_t)B * 512 * 128);
  float* l0new  = ALLOC((size_t)B * N0 * 128);
  int*   idx3a  = (int*)ALLOC((size_t)B * 512 * 3);
  float* w3a    = ALLOC((size_t)B * 512 * 3);
  int*   idx3b  = (int*)ALLOC((size_t)B * N0 * 3);
  float* w3b    = ALLOC((size_t)B * N0 * 3);
  float* stats  = ALLOC(2 * 1024);
  float* BUFA   = ALLOC((size_t)524288 * 96);   // X staging + odd intermediates
  float* BUFB   = ALLOC((size_t)524288 * 128);  // even intermediates / finals

  // ---- l0: transpose -------------------------------------------------------
  k_transpose<<<gblocks((size_t)B * N0 * 3), 256, 0, stream>>>(xyz_in, p0, B, N0);

  // ---- SA1 (MSG, npoint=512) ----------------------------------------------
  {
    size_t lds = (size_t)(4 * N0) * 4 + 256 * 4 + 256 * 4;
    k_fps<<<B, 256, lds, stream>>>(p0, fps1, N0, 512);
  }
  k_gather_xyz<<<gblocks((size_t)B * 512 * 3), 256, 0, stream>>>(p0, fps1, l1xyz, B, N0, 512);
  {
    const float r[3] = {0.1f, 0.2f, 0.4f};
    const int   K[3] = {32, 64, 128};
    const int   co[3] = {0, 64, 192};
    for (int br = 0; br < 3; ++br) {
      int S = 512, Cf = 3, cin = 6, cinPad = 16;
      k_ball_group<<<B * S, 128, 0, stream>>>(p0, p0, Cf, l1xyz, BUFA,
                                              N0, S, K[br], cinPad, r[br] * r[br]);
      int R = B * S * K[br];
      int c3 = sa1b[br][2].cout;
      run_chain(BUFA, R, cin, cinPad, sa1b[br], 3, BUFB, BUFA, BUFB, ru16(c3), stats, stream);
      k_maxpool<<<gblocks((size_t)B * S * c3), 256, 0, stream>>>(
          BUFB, l1pts, B * S, K[br], c3, ru16(c3), 320, co[br]);
    }
  }

  // ---- SA2 (MSG, npoint=128) ----------------------------------------------
  {
    int N1 = 512;
    size_t lds = (size_t)(4 * N1) * 4 + 256 * 4 + 256 * 4;
    k_fps<<<B, 256, lds, stream>>>(l1xyz, fps2, N1, 128);
    k_gather_xyz<<<gblocks((size_t)B * 128 * 3), 256, 0, stream>>>(l1xyz, fps2, l2xyz, B, N1, 128);
    const float r[2] = {0.4f, 0.8f};
    const int   K[2] = {64, 128};
    const int   co[2] = {0, 256};
    for (int br = 0; br < 2; ++br) {
      int S = 128, Cf = 320, cin = 323, cinPad = 336;
      k_ball_group<<<B * S, 128, 0, stream>>>(l1xyz, l1pts, Cf, l2xyz, BUFA,
                                              N1, S, K[br], cinPad, r[br] * r[br]);
      int R = B * S * K[br];
      run_chain(BUFA, R, cin, cinPad, sa2b[br], 3, BUFB, BUFA, BUFB, 256, stats, stream);
      k_maxpool<<<gblocks((size_t)B * S * 256), 256, 0, stream>>>(
          BUFB, l2pts, B * S, K[br], 256, 256, 512, co[br]);
    }
  }

  // ---- SA3 (group all) -----------------------------------------------------
  {
    int G = B * 128;
    k_groupall<<<gblocks((size_t)G * 528), 256, 0, stream>>>(l2xyz, l2pts, BUFA, G, 512, 528);
    run_chain(BUFA, G, 515, 528, sa3L, 3, BUFB, BUFA, BUFB, 1024, stats, stream);
    k_maxpool<<<gblocks((size_t)B * 1024), 256, 0, stream>>>(
        BUFB, l3pts, B, 128, 1024, 1024, 1024, 0);
  }

  // ---- FP3 (broadcast l3 -> l2) -------------------------------------------
  {
    int G = B * 128;
    k_fp3_stage<<<gblocks((size_t)G * 1536), 256, 0, stream>>>(l2pts, l3pts, BUFA, G, 128);
    run_chain(BUFA, G, 1536, 1536, fp3L, 2, BUFB, BUFA, l2new, 256, stats, stream);
  }

  // ---- FP2 (3-NN interp l2 -> l1) -----------------------------------------
  {
    int G = B * 512;
    k_knn3<<<(G + 255) / 256, 256, 0, stream>>>(l1xyz, l2xyz, idx3a, w3a, G, 512, 128);
    k_fp_stage<<<gblocks((size_t)G * 576), 256, 0, stream>>>(
        l1pts, 320, 320, l2new, 256, 256, idx3a, w3a, BUFA, G, 512, 128, 576);
    run_chain(BUFA, G, 576, 576, fp2L, 2, BUFB, BUFA, l1new, 128, stats, stream);
  }

  // ---- FP1 (3-NN interp l1 -> l0) -----------------------------------------
  {
    int G = B * N0;
    k_knn3<<<(G + 255) / 256, 256, 0, stream>>>(p0, l1xyz, idx3b, w3b, G, N0, 512);
    k_fp_stage<<<gblocks((size_t)G * 144), 256, 0, stream>>>(
        p0, 3, 3, l1new, 128, 128, idx3b, w3b, BUFA, G, N0, 512, 144);
    run_chain(BUFA, G, 131, 144, fp1L, 2, BUFB, BUFA, l0new, 128, stats, stream);
  }

  // ---- head: conv1 + BN + ReLU, conv2 -------------------------------------
  {
    int R = B * N0;
    LayerP c1 = {W1, b1, g1, be1, 128};
    launch_gemm(l0new, c1, BUFA, R, 128, 128, 128, stream);
    launch_bn(BUFA, R, 128, 128, g1, be1, stats, stream);
    LayerP c2 = {W2, b2, nullptr, nullptr, 128};
    launch_gemm(BUFA, c2, (float*)d_out, R, 128, 128, 128, stream);
  }
}